// Predictor_89713276878904
// MI455X (gfx1250) — compile-verified
//
#include <hip/hip_runtime.h>
#include <hip/hip_bf16.h>
#include <math.h>

typedef __attribute__((ext_vector_type(2))) float v2f;
typedef __attribute__((ext_vector_type(8))) float v8f;

constexpr int kN = 50000;   // nodes
constexpr int kE = 800000;  // edges
constexpr int kH = 64;      // feature / hidden dim
constexpr int kG = 64;      // graphs
constexpr int kC = 2;       // classes

// ---------------- small utility kernels ----------------

__global__ void zero_f32(float* p, int n) {
  int i = blockIdx.x * blockDim.x + threadIdx.x;
  if (i < n) p[i] = 0.0f;
}

__global__ void fill_u32(unsigned* p, unsigned v, int n) {
  int i = blockIdx.x * blockDim.x + threadIdx.x;
  if (i < n) p[i] = v;
}

__global__ void deg_accum(const long long* __restrict__ dst, float* __restrict__ deg, int e_cnt) {
  int e = blockIdx.x * blockDim.x + threadIdx.x;
  if (e < e_cnt) atomicAdd(&deg[dst[e]], 1.0f);
}

__global__ void deg_to_norm(float* deg_norm, int n) {
  int i = blockIdx.x * blockDim.x + threadIdx.x;
  if (i < n) deg_norm[i] = rsqrtf(deg_norm[i] + 1.0f);
}

// ---------------- fp32 WMMA GEMM: hout[N,64] = hin[N,64] @ W[64,64] ----------------
// One wave computes 16 rows x 64 cols. A: 16x4 fp32 tile (2 VGPR/lane),
// B: 4x16 tile, C/D: four 16x16 fp32 tiles (8 VGPR each).
// A layout (ISA 7.12.2): lane l holds M=l&15, K=(l>>4)*2+{0,1}.
// C/D layout: VGPR r, lanes 0-15 -> (M=r, N=lane); lanes 16-31 -> (M=8+r, N=lane-16).
__global__ void gemm_wmma(const float* __restrict__ hin,
                          const float* __restrict__ W,
                          float* __restrict__ hout, int nrows) {
  const int lane = threadIdx.x & 31;
  const int wave = threadIdx.x >> 5;
  const int row0 = (blockIdx.x * 4 + wave) * 16;
  if (row0 >= nrows) return;              // wave-uniform guard: EXEC stays all-ones
  const int m  = lane & 15;
  const int kb = (lane >> 4) << 1;        // 0 or 2
  v8f acc0 = {}, acc1 = {}, acc2 = {}, acc3 = {};
  const float* arow = hin + (long)(row0 + m) * kH;
#pragma unroll
  for (int k0 = 0; k0 < 64; k0 += 4) {
    v2f a;
    a.x = arow[k0 + kb];
    a.y = arow[k0 + kb + 1];
    const float* w0 = W + (long)(k0 + kb) * kH + m;   // row k0+kb, col base m
    const float* w1 = w0 + kH;                        // row k0+kb+1
    v2f b0 = { w0[0],  w1[0]  };
    v2f b1 = { w0[16], w1[16] };
    v2f b2 = { w0[32], w1[32] };
    v2f b3 = { w0[48], w1[48] };
    acc0 = __builtin_amdgcn_wmma_f32_16x16x4_f32(false, a, false, b0, (short)0, acc0, false, false);
    acc1 = __builtin_amdgcn_wmma_f32_16x16x4_f32(false, a, false, b1, (short)0, acc1, false, false);
    acc2 = __builtin_amdgcn_wmma_f32_16x16x4_f32(false, a, false, b2, (short)0, acc2, false, false);
    acc3 = __builtin_amdgcn_wmma_f32_16x16x4_f32(false, a, false, b3, (short)0, acc3, false, false);
  }
  const int om = (lane >> 4) << 3;        // 0 or 8
  const int on = lane & 15;
  float* orow = hout + (long)row0 * kH;
#pragma unroll
  for (int r = 0; r < 8; ++r) {
    float* p = orow + (long)(om + r) * kH + on;
    p[0]  = acc0[r];
    p[16] = acc1[r];
    p[32] = acc2[r];
    p[48] = acc3[r];
  }
}

// ---------------- edge scatter-add: agg[dst] += hw[src] * norm[src] ----------------
// 16 threads per edge, 4 floats each (float4 gather, 4 float atomics).
__global__ void edge_scatter(const float* __restrict__ hw,
                             const float* __restrict__ norm,
                             const long long* __restrict__ src,
                             const long long* __restrict__ dst,
                             float* __restrict__ agg, int e_cnt) {
  long tid = (long)blockIdx.x * blockDim.x + threadIdx.x;
  int e = (int)(tid >> 4);
  if (e >= e_cnt) return;
  int part = ((int)tid & 15) << 2;
  long s = src[e];
  long d = dst[e];
  float ns = norm[s];
  const float4 v = *(const float4*)(hw + s * 64 + part);
  float* ap = agg + d * 64 + part;
  atomicAdd(ap + 0, v.x * ns);
  atomicAdd(ap + 1, v.y * ns);
  atomicAdd(ap + 2, v.z * ns);
  atomicAdd(ap + 3, v.w * ns);
}

// ---------------- per-node finalize: out = agg*norm + hw*norm^2 + b (opt ReLU) ----------------
__global__ void layer_finalize(const float* __restrict__ agg,
                               const float* __restrict__ hw,
                               const float* __restrict__ norm,
                               const float* __restrict__ bias,
                               float* __restrict__ hout, int n_nodes, int relu) {
  long tid = (long)blockIdx.x * blockDim.x + threadIdx.x;
  if (tid >= (long)n_nodes * 64) return;
  int n = (int)(tid >> 6);
  int f = (int)tid & 63;
  float nn = norm[n];
  float v = agg[tid] * nn + hw[tid] * (nn * nn) + bias[f];
  if (relu) v = fmaxf(v, 0.0f);
  hout[tid] = v;
}

// ---------------- pooling (segment sum / max / count) ----------------
__device__ __forceinline__ unsigned enc_ordered(float f) {
  unsigned u = __float_as_uint(f);
  return (u & 0x80000000u) ? ~u : (u | 0x80000000u);
}
__device__ __forceinline__ float dec_ordered(unsigned e) {
  return (e & 0x80000000u) ? __uint_as_float(e & 0x7FFFFFFFu) : __uint_as_float(~e);
}

__global__ void pool_accum(const float* __restrict__ h,
                           const long long* __restrict__ batch,
                           float* __restrict__ psum,
                           unsigned* __restrict__ pmax,
                           float* __restrict__ pcnt, int n_nodes) {
  long tid = (long)blockIdx.x * blockDim.x + threadIdx.x;
  int n = (int)(tid >> 4);
  if (n >= n_nodes) return;
  int part = ((int)tid & 15) << 2;
  long g = batch[n];
  const float4 v = *(const float4*)(h + (long)n * 64 + part);
  float* sp = psum + g * 64 + part;
  unsigned* mp = pmax + g * 64 + part;
  atomicAdd(sp + 0, v.x); atomicMax(mp + 0, enc_ordered(v.x));
  atomicAdd(sp + 1, v.y); atomicMax(mp + 1, enc_ordered(v.y));
  atomicAdd(sp + 2, v.z); atomicMax(mp + 2, enc_ordered(v.z));
  atomicAdd(sp + 3, v.w); atomicMax(mp + 3, enc_ordered(v.w));
  if (part == 0) atomicAdd(&pcnt[g], 1.0f);
}

// ---------------- head: sigmoid(concat(mean,max) @ Wm + bm) ----------------
__global__ void head_kernel(const float* __restrict__ psum,
                            const unsigned* __restrict__ pmax,
                            const float* __restrict__ pcnt,
                            const float* __restrict__ Wm,
                            const float* __restrict__ bm,
                            float* __restrict__ out) {
  int t = threadIdx.x;
  if (t >= kG * kC) return;
  int g = t >> 1;
  int c = t & 1;
  float inv = 1.0f / fmaxf(pcnt[g], 1.0f);
  float acc = bm[c];
#pragma unroll 4
  for (int f = 0; f < 64; ++f)
    acc += psum[g * 64 + f] * inv * Wm[f * kC + c];
#pragma unroll 4
  for (int f = 0; f < 64; ++f)
    acc += dec_ordered(pmax[g * 64 + f]) * Wm[(64 + f) * kC + c];
  out[g * kC + c] = 1.0f / (1.0f + expf(-acc));
}

// ---------------- launch ----------------
extern "C" void kernel_launch(void* const* d_in, const int* in_sizes, int n_in,
                              void* d_out, int out_size, void* d_ws, size_t ws_size,
                              hipStream_t stream) {
  const float*     x     = (const float*)d_in[0];
  const long long* ei    = (const long long*)d_in[1];   // int64 [2,E]
  const long long* batch = (const long long*)d_in[2];   // int64 [N]
  const float* W0 = (const float*)d_in[3];  const float* b0 = (const float*)d_in[4];
  const float* W1 = (const float*)d_in[5];  const float* b1 = (const float*)d_in[6];
  const float* W2 = (const float*)d_in[7];  const float* b2 = (const float*)d_in[8];
  const float* Wm = (const float*)d_in[9];  const float* bm = (const float*)d_in[10];
  float* out = (float*)d_out;

  const long long* src = ei;
  const long long* dst = ei + kE;

  // workspace layout (floats), 64B-aligned chunks
  float* ws = (float*)d_ws;
  size_t off = 0;
  auto take = [&](size_t cnt) { float* p = ws + off; off += (cnt + 15) & ~size_t(15); return p; };
  float*    norm = take(kN);
  float*    hw   = take((size_t)kN * kH);
  float*    agg  = take((size_t)kN * kH);
  float*    hcur = take((size_t)kN * kH);
  float*    psum = take((size_t)kG * kH);
  float*    pcnt = take(kG);
  unsigned* pmax = (unsigned*)take((size_t)kG * kH);
  (void)ws_size; (void)in_sizes; (void)n_in; (void)out_size;

  const int B = 256;
  const int nNH = kN * kH;

  // degree (with self loop) -> norm = rsqrt(deg+1)
  zero_f32<<<(kN + B - 1) / B, B, 0, stream>>>(norm, kN);
  deg_accum<<<(kE + B - 1) / B, B, 0, stream>>>(dst, norm, kE);
  deg_to_norm<<<(kN + B - 1) / B, B, 0, stream>>>(norm, kN);

  const int gemmBlocks = (kN / 16 + 3) / 4;   // 4 waves (16 rows each) per 128-thread block
  const float* Ws[3] = {W0, W1, W2};
  const float* bs[3] = {b0, b1, b2};
  const float* hin = x;
  for (int l = 0; l < 3; ++l) {
    gemm_wmma<<<gemmBlocks, 128, 0, stream>>>(hin, Ws[l], hw, kN);
    zero_f32<<<(nNH + B - 1) / B, B, 0, stream>>>(agg, nNH);
    edge_scatter<<<(int)(((long)kE * 16 + B - 1) / B), B, 0, stream>>>(hw, norm, src, dst, agg, kE);
    layer_finalize<<<(nNH + B - 1) / B, B, 0, stream>>>(agg, hw, norm, bs[l], hcur, kN, l < 2 ? 1 : 0);
    hin = hcur;
  }

  // readout
  zero_f32<<<(kG * kH + B - 1) / B, B, 0, stream>>>(psum, kG * kH);
  zero_f32<<<1, B, 0, stream>>>(pcnt, kG);
  fill_u32<<<(kG * kH + B - 1) / B, B, 0, stream>>>(pmax, 0x007FFFFFu /* enc(-inf) */, kG * kH);
  pool_accum<<<(int)(((long)kN * 16 + B - 1) / B), B, 0, stream>>>(hcur, batch, psum, pmax, pcnt, kN);
  head_kernel<<<1, 128, 0, stream>>>(psum, pmax, pcnt, Wm, bm, out);
}